// Proposed_model_12893491822698
// MI455X (gfx1250) — compile-verified
//
#include <hip/hip_runtime.h>
#include <math.h>

typedef __attribute__((ext_vector_type(2))) float v2f;
typedef __attribute__((ext_vector_type(8))) float v8f;

#define FLAG_ADDC 1
#define FLAG_BIAS 2
#define FLAG_RELU 4

struct AttnArgs {
  const float* t[7];
  const float* p[7];
  float scale[7];
};

__device__ __forceinline__ v8f wmma_f32_16x16x4(v2f a, v2f b, v8f c) {
#if defined(__AMDGCN__) && __has_builtin(__builtin_amdgcn_wmma_f32_16x16x4_f32)
  // 8 args: (neg_a, A, neg_b, B, c_mod, C, reuse_a, reuse_b)
  return __builtin_amdgcn_wmma_f32_16x16x4_f32(false, a, false, b, (short)0, c,
                                               false, false);
#else
  // Host-pass / missing-builtin placeholder (never the intended device path).
  c[0] += a.x * b.x + a.y * b.y;
  return c;
#endif
}

// C[M,N] = A[M,K] @ B[K,N] (+C)(+bias)(relu). Row-major. M%16==0, N%64==0, K%64==0.
// Block = 256 threads = 8 waves; each wave computes a 16x64 tile.
// B panel (64K x 64N chunk, 16 KB) is cooperatively staged in LDS and shared by
// all 8 waves; inner loop: ds_load b-fragments + v_wmma_f32_16x16x4_f32.
__global__ __launch_bounds__(256) void wmma_gemm_kernel(
    const float* __restrict__ A, const float* __restrict__ B,
    const float* __restrict__ bias, const float* __restrict__ Cadd,
    float* __restrict__ C, int M, int N, int K, int flags) {
  __shared__ float Bs[64 * 64];  // 16 KB

  const int lane = threadIdx.x & 31;
  const int wave = threadIdx.x >> 5;
  const int lo = lane & 15;
  const int hi = lane >> 4;
  const int mbase = blockIdx.y * 128 + wave * 16;
  const bool active = (mbase < M);  // inactive waves still join barriers/staging
  const int nbase = blockIdx.x * 64;

  v8f acc[4] = {};
  const float* Arow = A + (size_t)((active ? mbase : 0) + lo) * K;

  for (int kc = 0; kc < K; kc += 64) {
    // Stage B[kc..kc+63][nbase..nbase+63] into LDS: 1024 float4s / 256 threads.
#pragma unroll
    for (int i = 0; i < 4; ++i) {
      const int idx = threadIdx.x + i * 256;  // one float4 each
      const int r = idx >> 4;                 // 16 float4 per 64-col row
      const int c4 = (idx & 15) * 4;
      *(float4*)&Bs[r * 64 + c4] =
          *(const float4*)&B[(size_t)(kc + r) * N + nbase + c4];
    }
    __syncthreads();

    for (int k0 = 0; k0 < 64; k0 += 4) {
      const int ka = k0 + 2 * hi;  // lanes 0-15: K=k0,k0+1 ; lanes 16-31: K=k0+2,k0+3
      const float2 av = *(const float2*)(Arow + kc + ka);
      v2f a;
      a.x = av.x;
      a.y = av.y;
#pragma unroll
      for (int c = 0; c < 4; ++c) {
        v2f b;
        b.x = Bs[ka * 64 + c * 16 + lo];
        b.y = Bs[(ka + 1) * 64 + c * 16 + lo];
        acc[c] = wmma_f32_16x16x4(a, b, acc[c]);
      }
    }
    __syncthreads();
  }

  if (!active) return;
#pragma unroll
  for (int c = 0; c < 4; ++c) {
    const int n = nbase + c * 16 + lo;
    const float bv = (flags & FLAG_BIAS) ? bias[n] : 0.0f;
#pragma unroll
    for (int r = 0; r < 8; ++r) {
      const int m = mbase + r + 8 * hi;  // C layout: vgpr r -> rows r / r+8
      const size_t off = (size_t)m * N + n;
      float v = acc[c][r] + bv;
      if (flags & FLAG_ADDC) v += Cadd[off];
      if (flags & FLAG_RELU) v = fmaxf(v, 0.0f);
      C[off] = v;
    }
  }
}

// One wave per edge; lane p handles float4 chunk p of the 128-float row.
// accum must be pre-zeroed. coef = rsq_src[s] * (ew1[e]+ew2[e] if weighted).
__global__ __launch_bounds__(256) void edge_scatter_kernel(
    const float* __restrict__ hsrc, const int* __restrict__ src,
    const int* __restrict__ dst, const float* __restrict__ rsq_src,
    const float* __restrict__ ew1, const float* __restrict__ ew2,
    float* __restrict__ accum, int E) {
  const long long gid = (long long)blockIdx.x * blockDim.x + threadIdx.x;
  const int e = (int)(gid >> 5);
  if (e >= E) return;
  const int p = (int)(gid & 31);
  const int s = src[e];  // wave-uniform load
  const int d = dst[e];
  float coef = rsq_src[s];
  if (ew1 != nullptr) coef *= (ew1[e] + ew2[e]);
  const float4 v = *(const float4*)(hsrc + (size_t)s * 128 + p * 4);
  float* dp = accum + (size_t)d * 128 + p * 4;
  atomicAdd(dp + 0, v.x * coef);
  atomicAdd(dp + 1, v.y * coef);
  atomicAdd(dp + 2, v.z * coef);
  atomicAdd(dp + 3, v.w * coef);
}

__global__ void count_kernel(const int* __restrict__ idx, float* __restrict__ cnt,
                             int n) {
  const int g = blockIdx.x * blockDim.x + threadIdx.x;
  if (g < n) atomicAdd(&cnt[idx[g]], 1.0f);
}

__global__ void rsqrt_clamp_kernel(float* __restrict__ c, int n) {
  const int g = blockIdx.x * blockDim.x + threadIdx.x;
  if (g < n) c[g] = rsqrtf(fmaxf(c[g], 1.0f));
}

// h[row, :] *= w[row]  (feature dim fixed at 128)
__global__ void row_scale_kernel(float* __restrict__ h, const float* __restrict__ w,
                                 long long total) {
  const long long g = (long long)blockIdx.x * blockDim.x + threadIdx.x;
  if (g < total) h[g] *= w[g >> 7];
}

// out[n,j] = sum_l softmax_l( scale_l * P_l[n,j] * a[j] ) * scale_l * t_l[n,j]
__global__ void attn_combine_kernel(AttnArgs args, const float* __restrict__ a,
                                    float* __restrict__ out, int total, int L) {
  const int g = blockIdx.x * blockDim.x + threadIdx.x;
  if (g >= total) return;
  const float av = a[g & 127];
  float lg[7];
  float m = -3.0e38f;
  for (int l = 0; l < L; ++l) {
    const float v = args.p[l][g] * args.scale[l] * av;
    lg[l] = v;
    m = fmaxf(m, v);
  }
  float den = 0.0f, num = 0.0f;
  for (int l = 0; l < L; ++l) {
    const float w = expf(lg[l] - m);
    den += w;
    num += w * args.t[l][g] * args.scale[l];
  }
  out[g] = num / den;
}

__global__ void combine_game_kernel(const float* __restrict__ hand,
                                    const float* __restrict__ hor,
                                    const float* __restrict__ hg,
                                    float* __restrict__ out, int total) {
  const int g = blockIdx.x * blockDim.x + threadIdx.x;
  if (g >= total)
    return;
  out[g] = 0.25f * hand[g] + 0.5f * hor[g] + 0.25f * hg[g];  // AND/OR/SELF weights
}

static inline int cdiv_i(long long a, long long b) { return (int)((a + b - 1) / b); }

extern "C" void kernel_launch(void* const* d_in, const int* in_sizes, int n_in,
                              void* d_out, int out_size, void* d_ws, size_t ws_size,
                              hipStream_t stream) {
  (void)in_sizes; (void)n_in; (void)out_size;
  constexpr int U_N = 100000, I_N = 50000, DIM = 128, LLMD = 768;
  constexpr int E_UI = 1000000, E_AND = 500000, E_OR = 500000;

  const float* user_emb = (const float*)d_in[0];
  const float* item_emb = (const float*)d_in[1];
  const float* user_llm = (const float*)d_in[2];
  const float* item_llm = (const float*)d_in[3];
  const float* W_and = (const float*)d_in[4];
  const float* a_and = (const float*)d_in[5];
  const float* W_or = (const float*)d_in[6];
  const float* a_or = (const float*)d_in[7];
  const float* uf_w1 = (const float*)d_in[8];
  const float* uf_b1 = (const float*)d_in[9];
  const float* uf_w2 = (const float*)d_in[10];
  const float* uf_b2 = (const float*)d_in[11];
  const float* il_w1 = (const float*)d_in[12];
  const float* il_b1 = (const float*)d_in[13];
  const float* il_w2 = (const float*)d_in[14];
  const float* il_b2 = (const float*)d_in[15];
  const float* if_w1 = (const float*)d_in[16];
  const float* if_b1 = (const float*)d_in[17];
  const float* if_w2 = (const float*)d_in[18];
  const float* if_b2 = (const float*)d_in[19];
  const float* w_penr = (const float*)d_in[20];
  const float* w_per = (const float*)d_in[21];
  const float* w_node = (const float*)d_in[22];
  const int* ui_user = (const int*)d_in[23];
  const int* ui_game = (const int*)d_in[24];
  const int* as[3] = {(const int*)d_in[25], (const int*)d_in[27], (const int*)d_in[29]};
  const int* ad[3] = {(const int*)d_in[26], (const int*)d_in[28], (const int*)d_in[30]};
  const int* ors = (const int*)d_in[31];
  const int* ord_ = (const int*)d_in[32];

  // ---- workspace layout (floats) ----
  float* wsf = (float*)d_ws;
  size_t off = 0;
  auto alloc = [&](size_t n) { float* p = wsf + off; off += n; return p; };
  float* c_uu = alloc(U_N);
  float* c_ug = alloc(I_N);
  float* c_gs[3];
  float* c_gd[3];
  for (int g = 0; g < 3; ++g) { c_gs[g] = alloc(I_N); c_gd[g] = alloc(I_N); }
  float* c_os = alloc(I_N);
  float* c_od = alloc(I_N);
  const size_t cnt_total = off;  // U + 9I, contiguous
  float* Buser = alloc((size_t)U_N * DIM);
  float* Cuser = alloc((size_t)U_N * DIM);
  float* Dgame = alloc((size_t)I_N * DIM);
  float* Egame = alloc((size_t)I_N * DIM);
  float* Ftmp = alloc((size_t)I_N * 64);
  float* Gllm = alloc((size_t)I_N * DIM);
  float* Hhid = alloc((size_t)U_N * DIM);
  float* T[6];
  for (int i = 0; i < 6; ++i) T[i] = alloc((size_t)I_N * DIM);
  float* O1 = alloc((size_t)I_N * DIM);
  float* O2 = alloc((size_t)I_N * DIM);
  if (off * sizeof(float) > ws_size) return;  // workspace too small

  float* out = (float*)d_out;
  float* o_user = out;
  float* o_game = out + (size_t)U_N * DIM;
  float* o_and = out + (size_t)(U_N + I_N) * DIM;
  float* o_or = out + (size_t)(U_N + 2 * I_N) * DIM;

  auto gemm = [&](const float* A, const float* B, const float* bias,
                  const float* Cadd, float* C, int M, int N, int K, int flags) {
    dim3 grid(N / 64, (M + 127) / 128);
    wmma_gemm_kernel<<<grid, 256, 0, stream>>>(A, B, bias, Cadd, C, M, N, K, flags);
  };
  auto rowscale = [&](float* h, const float* w, long long total) {
    row_scale_kernel<<<cdiv_i(total, 256), 256, 0, stream>>>(h, w, total);
  };
  auto conv = [&](const float* hsrc, const int* src, const int* dst,
                  const float* rsq_s, const float* rsq_d, const float* e1,
                  const float* e2, float* outb, int E, size_t outFloats) {
    hipMemsetAsync(outb, 0, outFloats * sizeof(float), stream);
    edge_scatter_kernel<<<cdiv_i((long long)E * 32, 256), 256, 0, stream>>>(
        hsrc, src, dst, rsq_s, e1, e2, outb, E);
    rowscale(outb, rsq_d, (long long)outFloats);
  };
  auto count = [&](const int* idx, float* cnt, int n) {
    count_kernel<<<cdiv_i(n, 256), 256, 0, stream>>>(idx, cnt, n);
  };

  // ---- degrees -> rsqrt(max(deg,1)) ----
  hipMemsetAsync(c_uu, 0, cnt_total * sizeof(float), stream);
  count(ui_user, c_uu, E_UI);
  count(ui_game, c_ug, E_UI);
  for (int g = 0; g < 3; ++g) {
    count(as[g], c_gs[g], E_AND);
    count(ad[g], c_gd[g], E_AND);
  }
  count(ors, c_os, E_OR);
  count(ord_, c_od, E_OR);
  rsqrt_clamp_kernel<<<cdiv_i((long long)cnt_total, 256), 256, 0, stream>>>(
      c_uu, (int)cnt_total);

  // ---- LLM fusion MLPs (WMMA GEMMs; concat handled as accumulating GEMM) ----
  gemm(item_llm, il_w1, il_b1, nullptr, Ftmp, I_N, 64, LLMD, FLAG_BIAS | FLAG_RELU);
  gemm(Ftmp, il_w2, il_b2, nullptr, Gllm, I_N, DIM, 64, FLAG_BIAS);
  gemm(user_emb, uf_w1, nullptr, nullptr, Hhid, U_N, DIM, DIM, 0);
  gemm(user_llm, uf_w1 + DIM * DIM, uf_b1, Hhid, Hhid, U_N, DIM, DIM,
       FLAG_ADDC | FLAG_BIAS | FLAG_RELU);
  gemm(Hhid, uf_w2, uf_b2, nullptr, Buser, U_N, DIM, DIM, FLAG_BIAS);
  gemm(item_emb, if_w1, nullptr, nullptr, Hhid, I_N, DIM, DIM, 0);
  gemm(Gllm, if_w1 + DIM * DIM, if_b1, Hhid, Hhid, I_N, DIM, DIM,
       FLAG_ADDC | FLAG_BIAS | FLAG_RELU);
  gemm(Hhid, if_w2, if_b2, nullptr, Dgame, I_N, DIM, DIM, FLAG_BIAS);

  // ---- user <-> game propagation (2 rounds) ----
  float* ucur = Buser;
  float* unxt = Cuser;
  float* gcur = Dgame;
  float* gnxt = Egame;
  for (int it = 0; it < 2; ++it) {
    rowscale(gcur, w_node, (long long)I_N * DIM);  // diag(weight_node_PENR) @ h_game
    conv(gcur, ui_game, ui_user, c_ug, c_uu, w_penr, w_per, unxt, E_UI,
         (size_t)U_N * DIM);  // game -> user, PENR+PER weighted
    conv(ucur, ui_user, ui_game, c_uu, c_ug, nullptr, nullptr, gnxt, E_UI,
         (size_t)I_N * DIM);  // old user -> game, unweighted
    float* t = ucur; ucur = unxt; unxt = t;
    t = gcur; gcur = gnxt; gnxt = t;
  }
  hipMemcpyAsync(o_user, ucur, (size_t)U_N * DIM * sizeof(float),
                 hipMemcpyDeviceToDevice, stream);

  // ---- AND subgraphs: 3 graphs x 2 layers ----
  for (int g = 0; g < 3; ++g)
    conv(item_emb, as[g], ad[g], c_gs[g], c_gd[g], nullptr, nullptr, T[g], E_AND,
         (size_t)I_N * DIM);
  for (int g = 0; g < 3; ++g)
    conv(T[g], as[g], ad[g], c_gs[g], c_gd[g], nullptr, nullptr, T[3 + g], E_AND,
         (size_t)I_N * DIM);
  // ---- OR chain: 2 layers ----
  conv(item_emb, ors, ord_, c_os, c_od, nullptr, nullptr, O1, E_OR,
       (size_t)I_N * DIM);
  conv(O1, ors, ord_, c_os, c_od, nullptr, nullptr, O2, E_OR, (size_t)I_N * DIM);

  // ---- layer attention (logits P_l = t_l @ W via WMMA, fused softmax-combine)
  // P buffers alias dead ping-pong/MLP buffers + the not-yet-written h_or output.
  float* P[7] = {Cuser, Cuser + (size_t)I_N * DIM, Hhid, Hhid + (size_t)I_N * DIM,
                 Gllm, Egame, o_or};
  const float* tAnd[7] = {item_emb, T[0], T[1], T[2], T[3], T[4], T[5]};
  for (int l = 0; l < 7; ++l)
    gemm(tAnd[l], W_and, nullptr, nullptr, P[l], I_N, DIM, DIM, 0);
  {
    AttnArgs aa = {};
    for (int l = 0; l < 7; ++l) { aa.t[l] = tAnd[l]; aa.p[l] = P[l]; aa.scale[l] = 1.0f; }
    attn_combine_kernel<<<cdiv_i((long long)I_N * DIM, 256), 256, 0, stream>>>(
        aa, a_and, o_and, I_N * DIM, 7);
  }
  const float* tOr[3] = {item_emb, O1, O2};
  const float scOr[3] = {1.0f, 0.9f, 1.0f};  // depth-decay params (li=1: 0.9, li=2: 1.0)
  for (int l = 0; l < 3; ++l)
    gemm(tOr[l], W_or, nullptr, nullptr, P[l], I_N, DIM, DIM, 0);
  {
    AttnArgs ao = {};
    for (int l = 0; l < 3; ++l) { ao.t[l] = tOr[l]; ao.p[l] = P[l]; ao.scale[l] = scOr[l]; }
    attn_combine_kernel<<<cdiv_i((long long)I_N * DIM, 256), 256, 0, stream>>>(
        ao, a_or, o_or, I_N * DIM, 3);
  }

  // ---- final h_game = 0.25*h_and + 0.5*h_or + 0.25*h_game_prop ----
  combine_game_kernel<<<cdiv_i((long long)I_N * DIM, 256), 256, 0, stream>>>(
      o_and, o_or, gcur, o_game, I_N * DIM);
}